// GraphSAGENet_25460566130853
// MI455X (gfx1250) — compile-verified
//
#include <hip/hip_runtime.h>

#define N_NODES 50000
#define N_EDGES 800000
#define DIM 96
#define WPAD 98   // LDS row stride (floats): l*98*? -> lane bank l*34 mod 64, conflict-free

typedef __attribute__((ext_vector_type(2))) float v2f;
typedef __attribute__((ext_vector_type(8))) float v8f;

// One wave per edge: gather 96-float source row (coalesced), scatter-add into
// destination row with global float atomics (L2-resident target: 19.2MB << 192MB L2).
__global__ __launch_bounds__(256) void scatter_kernel(
    const float* __restrict__ h, const int* __restrict__ ei,
    float* __restrict__ agg, float* __restrict__ deg,
    int E, int addDeg)
{
  int gid  = blockIdx.x * 256 + threadIdx.x;
  int e    = gid >> 5;
  int lane = threadIdx.x & 31;
  if (e >= E) return;
  int src = ei[e];
  int dst = ei[E + e];
  const float* hs = h   + (size_t)src * DIM;
  float*       ad = agg + (size_t)dst * DIM;
  float v0 = hs[lane];
  float v1 = hs[lane + 32];
  float v2 = hs[lane + 64];
  atomicAdd(ad + lane,      v0);
  atomicAdd(ad + lane + 32, v1);
  atomicAdd(ad + lane + 64, v2);
  if (addDeg && lane == 0) atomicAdd(deg + dst, 1.0f);
}

__global__ __launch_bounds__(256) void rdeg_kernel(float* deg, int n) {
  int i = blockIdx.x * 256 + threadIdx.x;
  if (i < n) deg[i] = 1.0f / fmaxf(deg[i], 1.0f);
}

// Fused SAGE layer GEMM: out = (summed * rdeg) @ Wl^T + bl + h @ Wr^T  [+ReLU]
// One wave per 16-node tile; native fp32 WMMA 16x16x4, K=96 (24 steps), 6 j-tiles.
// Wl/Wr staged in LDS once per block (75KB of the WGP's 320KB), padded stride
// for conflict-free ds_load_b64 on the B operands.
__global__ __launch_bounds__(256) void sage_wmma_kernel(
    const float* __restrict__ summed, const float* __restrict__ rdeg,
    const float* __restrict__ h,
    const float* __restrict__ Wl, const float* __restrict__ bl,
    const float* __restrict__ Wr,
    float* __restrict__ out, int ntiles, int doRelu)
{
  __shared__ float sWl[DIM * WPAD];
  __shared__ float sWr[DIM * WPAD];

  // Cooperative stage of both 96x96 weight matrices (all 256 threads).
  for (int idx = threadIdx.x; idx < DIM * DIM; idx += 256) {
    int r = idx / DIM, c = idx - r * DIM;
    sWl[r * WPAD + c] = Wl[idx];
    sWr[r * WPAD + c] = Wr[idx];
  }
  __syncthreads();

  int tile = blockIdx.x * (blockDim.x >> 5) + (threadIdx.x >> 5);
  if (tile < ntiles) {                   // wave-uniform: EXEC all-1s inside
    int lane = threadIdx.x & 31;
    int li   = lane & 15;                // A: M row index; B: N col index
    int half = lane >> 4;                // selects K offset {0,2} per ISA layout
    int row0 = tile * 16;
    int row  = row0 + li;

    float rd = rdeg[row];
    const float* srow = summed + (size_t)row * DIM;
    const float* hrow = h      + (size_t)row * DIM;

    v8f acc[6];
#pragma unroll
    for (int jt = 0; jt < 6; ++jt)
#pragma unroll
      for (int r = 0; r < 8; ++r) acc[jt][r] = 0.0f;

    for (int kb = 0; kb < DIM; kb += 4) {
      int k0 = kb + half * 2;            // even -> 8B-aligned v2f loads
      v2f s  = *(const v2f*)(srow + k0);
      v2f a1; a1.x = s.x * rd; a1.y = s.y * rd;   // fused mean division
      v2f a2 = *(const v2f*)(hrow + k0);
#pragma unroll
      for (int jt = 0; jt < 6; ++jt) {
        int j = jt * 16 + li;            // B[k][j] = W[j][k] (W^T on the fly)
        v2f bL = *(const v2f*)(sWl + j * WPAD + k0);   // ds_load_b64, no conflicts
        v2f bR = *(const v2f*)(sWr + j * WPAD + k0);
        acc[jt] = __builtin_amdgcn_wmma_f32_16x16x4_f32(
            false, a1, false, bL, (short)0, acc[jt], false, false);
        acc[jt] = __builtin_amdgcn_wmma_f32_16x16x4_f32(
            false, a2, false, bR, (short)0, acc[jt], false, false);
      }
    }

#pragma unroll
    for (int jt = 0; jt < 6; ++jt) {
      float bj = bl[jt * 16 + li];
#pragma unroll
      for (int r = 0; r < 8; ++r) {
        float v = acc[jt][r] + bj;
        if (doRelu) v = fmaxf(v, 0.0f);
        // C/D layout: VGPR r holds rows {r (lanes 0-15), r+8 (lanes 16-31)}
        out[(size_t)(row0 + r + half * 8) * DIM + jt * 16 + li] = v;
      }
    }
  }
}

extern "C" void kernel_launch(void* const* d_in, const int* in_sizes, int n_in,
                              void* d_out, int out_size, void* d_ws, size_t ws_size,
                              hipStream_t stream) {
  const float* x   = (const float*)d_in[0];
  const int*   ei  = (const int*)  d_in[1];   // edge_index [2, E] (JAX default x64-off -> int32)
  const float* Wl1 = (const float*)d_in[2];
  const float* bl1 = (const float*)d_in[3];
  const float* Wr1 = (const float*)d_in[4];
  const float* Wl2 = (const float*)d_in[5];
  const float* bl2 = (const float*)d_in[6];
  const float* Wr2 = (const float*)d_in[7];
  float* out = (float*)d_out;

  float* agg = (float*)d_ws;                         // N*96 floats (19.2 MB)
  float* deg = agg + (size_t)N_NODES * DIM;          // N floats (becomes rdeg)
  float* h1  = deg + N_NODES;                        // N*96 floats

  int scatterBlocks = (N_EDGES * 32) / 256;          // one wave per edge, 8 waves/block
  int ntiles        = N_NODES / 16;                  // 3125 exact tiles
  int gemmBlocks    = (ntiles + 7) / 8;

  // ---- Layer 1 ----
  hipMemsetAsync(agg, 0, (size_t)N_NODES * DIM * sizeof(float), stream);
  hipMemsetAsync(deg, 0, (size_t)N_NODES * sizeof(float), stream);
  scatter_kernel<<<scatterBlocks, 256, 0, stream>>>(x, ei, agg, deg, N_EDGES, 1);
  rdeg_kernel<<<(N_NODES + 255) / 256, 256, 0, stream>>>(deg, N_NODES);
  sage_wmma_kernel<<<gemmBlocks, 256, 0, stream>>>(agg, deg, x, Wl1, bl1, Wr1,
                                                   h1, ntiles, 1);
  // ---- Layer 2 ----
  hipMemsetAsync(agg, 0, (size_t)N_NODES * DIM * sizeof(float), stream);
  scatter_kernel<<<scatterBlocks, 256, 0, stream>>>(h1, ei, agg, deg, N_EDGES, 0);
  sage_wmma_kernel<<<gemmBlocks, 256, 0, stream>>>(agg, deg, h1, Wl2, bl2, Wr2,
                                                   out, ntiles, 0);
}